// NetVLADLayer_33552284516975
// MI455X (gfx1250) — compile-verified
//
#include <hip/hip_runtime.h>
#include <hip/hip_bf16.h>
#include <math.h>

#define NN 32
#define CC 512
#define KK 64
#define PP 1024
#define ALPHAC 100.0f
#define EPSN 1e-12f
#define BSTRIDE 40   // LDS row pitch (floats): rows c,c+2 differ by 80 = 16 mod 64 banks -> conflict-free

typedef __attribute__((ext_vector_type(2))) float v2f;
typedef __attribute__((ext_vector_type(8))) float v8f;
typedef __attribute__((address_space(3))) float lds_f;

// ---------------------------------------------------------------- bias[k] = -alpha*||vocab_k||
__global__ void nv_bias_kernel(const float* __restrict__ vocabs, float* __restrict__ bias) {
  __shared__ float red[256];
  const int k = blockIdx.x;
  float s = 0.f;
  for (int c = threadIdx.x; c < CC; c += 256) { float v = vocabs[k * CC + c]; s += v * v; }
  red[threadIdx.x] = s; __syncthreads();
  for (int off = 128; off > 0; off >>= 1) {
    if (threadIdx.x < off) red[threadIdx.x] += red[threadIdx.x + off];
    __syncthreads();
  }
  if (threadIdx.x == 0) bias[k] = -ALPHAC * sqrtf(red[0]);
}

// ---------------------------------------------------------------- invn[n,p] = 1/max(||x[n,:,p]||, eps)
__global__ void nv_invnorm_kernel(const float* __restrict__ x, float* __restrict__ invn) {
  const int t = blockIdx.x * 256 + threadIdx.x;   // over N*P
  const int n = t >> 10, p = t & (PP - 1);
  const float* xp = x + (size_t)n * CC * PP + p;
  float s = 0.f;
  for (int c = 0; c < CC; ++c) { float v = xp[(size_t)c * PP]; s += v * v; }
  invn[t] = 1.0f / fmaxf(sqrtf(s), EPSN);
}

// ---------------------------------------------------------------- GEMM1 (f32 WMMA) + fused softmax over K
// block = 256 thr (8 waves), grid = N * (P/32). B panel (512c x 32p) staged once per block into LDS
// with GLOBAL_LOAD_ASYNC_TO_LDS_B128 (ASYNCcnt), removing the 4x duplicated VMEM reads across k-tiles.
__global__ void __launch_bounds__(256) nv_gemm1_softmax_kernel(
    const float* __restrict__ x, const float* __restrict__ vocabs,
    const float* __restrict__ invn, const float* __restrict__ bias,
    float* __restrict__ assign) {
  __shared__ float smB[CC][BSTRIDE];               // 512 x 40 floats = 80 KB
  __shared__ float sm[KK][33];                     // softmax staging: 64 k x 32 p (padded)
  const int n      = blockIdx.x >> 5;
  const int ptile0 = (blockIdx.x & 31) << 5;
  const int tid  = threadIdx.x;
  const int w    = tid >> 5;
  const int lane = tid & 31;
  const int half = lane >> 4;
  const int idx  = lane & 15;
  const int kt = w & 3;                            // k-tile 0..3
  const int pt = w >> 2;                           // p-subtile 0..1
  const int p0 = ptile0 + pt * 16;

  // ---- async stage: 512 rows x 128B; one B128 per lane => 256 lanes x 16B = 32 rows / instr
  {
    const int rowInGrp = tid >> 3;                 // 0..31
    const int chunk    = tid & 7;                  // 0..7  (16B chunks in a 128B row)
    const char* gbase = (const char*)(x + (size_t)n * CC * PP + ptile0);
#pragma unroll
    for (int i = 0; i < 16; ++i) {
      const int c = i * 32 + rowInGrp;
      const void* gaddr = gbase + (size_t)c * (PP * 4) + chunk * 16;
      const uint32_t lds_off = (uint32_t)(size_t)(lds_f*)(&smB[c][chunk * 4]);
      asm volatile("global_load_async_to_lds_b128 %0, %1, off"
                   :: "v"(lds_off), "v"(gaddr) : "memory");
    }
    asm volatile("s_wait_asynccnt 0x0" ::: "memory");
  }
  __syncthreads();

  const float* Arow = vocabs + (size_t)(kt * 16 + idx) * CC;   // A[m=idx][c]
  const float  binv = invn[n * PP + p0 + idx];
  const int    col  = pt * 16 + idx;

  v8f acc = {};
  for (int c0 = 0; c0 < CC; c0 += 4) {
    v2f a; a.x = Arow[c0 + 2 * half + 0] * (2.0f * ALPHAC);
           a.y = Arow[c0 + 2 * half + 1] * (2.0f * ALPHAC);
    v2f b; b.x = smB[c0 + 2 * half + 0][col] * binv;
           b.y = smB[c0 + 2 * half + 1][col] * binv;
    acc = __builtin_amdgcn_wmma_f32_16x16x4_f32(false, a, false, b, (short)0, acc, false, false);
  }
#pragma unroll
  for (int v = 0; v < 8; ++v) {
    const int kk2 = kt * 16 + v + 8 * half;        // D row -> cluster index
    sm[kk2][pt * 16 + idx] = acc[v] + bias[kk2];
  }
  __syncthreads();
  if (tid < 32) {                                  // softmax over K per p-column
    const int pl = tid;
    float mx = -3.0e38f;
    for (int kk2 = 0; kk2 < KK; ++kk2) mx = fmaxf(mx, sm[kk2][pl]);
    float ssum = 0.f;
    for (int kk2 = 0; kk2 < KK; ++kk2) { float e = __expf(sm[kk2][pl] - mx); sm[kk2][pl] = e; ssum += e; }
    const float r = 1.0f / ssum;
    for (int kk2 = 0; kk2 < KK; ++kk2) sm[kk2][pl] *= r;
  }
  __syncthreads();
  for (int i = tid; i < KK * 32; i += 256) {
    const int kk2 = i >> 5, pl = i & 31;
    assign[((size_t)n * KK + kk2) * PP + ptile0 + pl] = sm[kk2][pl];
  }
}

// ---------------------------------------------------------------- GEMM2 (f32 WMMA): vlad[n,k,c]
// tile = 16(k) x 16(c), inner = P; A row-sum (Sigma_p a) folded in via wave32 shfl reduction.
__global__ void __launch_bounds__(256) nv_gemm2_vlad_kernel(
    const float* __restrict__ x, const float* __restrict__ assign,
    const float* __restrict__ invn, const float* __restrict__ vocabs,
    float* __restrict__ vladb) {
  const int w    = threadIdx.x >> 5;
  const int lane = threadIdx.x & 31;
  const int half = lane >> 4;
  const int idx  = lane & 15;
  const int t     = blockIdx.x * 8 + w;            // 4096 tiles total
  const int n     = t >> 7;
  const int rem   = t & 127;
  const int ktile = rem >> 5;                      // 0..3
  const int ctile = rem & 31;                      // 0..31

  const float* Arow = assign + ((size_t)n * KK + ktile * 16 + idx) * PP;   // a[n][k=..][p]
  const float* Bcol = x + ((size_t)n * CC + ctile * 16 + idx) * PP;        // x[n][c=..][p]
  const float* ivn  = invn + n * PP;

  v8f acc = {};
  float lsum = 0.f;
  for (int p0 = 0; p0 < PP; p0 += 4) {
    v2f a; a.x = Arow[p0 + 2 * half + 0];
           a.y = Arow[p0 + 2 * half + 1];
    lsum += a.x + a.y;                             // builds Sigma_p a[k,p] per lane
    v2f b; b.x = Bcol[p0 + 2 * half + 0] * ivn[p0 + 2 * half + 0];
           b.y = Bcol[p0 + 2 * half + 1] * ivn[p0 + 2 * half + 1];
    acc = __builtin_amdgcn_wmma_f32_16x16x4_f32(false, a, false, b, (short)0, acc, false, false);
  }
  const float asum = lsum + __shfl_xor(lsum, 16, 32);  // lanes idx & idx+16 -> full row sum for k=idx
#pragma unroll
  for (int v = 0; v < 8; ++v) {
    const int m = v + 8 * half;
    const float as_m = __shfl(asum, m, 32);        // broadcast row-m sum from lane m
    const int kk2 = ktile * 16 + m;
    const int cc2 = ctile * 16 + idx;
    vladb[((size_t)n * KK + kk2) * CC + cc2] = acc[v] - as_m * vocabs[kk2 * CC + cc2];
  }
}

// ---------------------------------------------------------------- intra-norm stats per (n,k) row
__global__ void nv_rownorm_kernel(const float* __restrict__ vladb,
                                  float* __restrict__ rinv, float* __restrict__ partial) {
  __shared__ float red[256];
  const int nk = blockIdx.x;
  const float* row = vladb + (size_t)nk * CC;
  float s = 0.f;
  for (int c = threadIdx.x; c < CC; c += 256) { float v = row[c]; s += v * v; }
  red[threadIdx.x] = s; __syncthreads();
  for (int off = 128; off > 0; off >>= 1) {
    if (threadIdx.x < off) red[threadIdx.x] += red[threadIdx.x + off];
    __syncthreads();
  }
  if (threadIdx.x == 0) {
    const float ri = 1.0f / fmaxf(sqrtf(red[0]), EPSN);
    rinv[nk] = ri;
    partial[nk] = red[0] * ri * ri;                // ||row/max(||row||,eps)||^2
  }
}

// ---------------------------------------------------------------- final L2 per image
__global__ void nv_finalnorm_kernel(const float* __restrict__ partial, float* __restrict__ finv) {
  __shared__ float red[64];
  const int n = blockIdx.x;
  red[threadIdx.x] = partial[n * KK + threadIdx.x];
  __syncthreads();
  for (int off = 32; off > 0; off >>= 1) {
    if (threadIdx.x < off) red[threadIdx.x] += red[threadIdx.x + off];
    __syncthreads();
  }
  if (threadIdx.x == 0) finv[n] = 1.0f / fmaxf(sqrtf(red[0]), EPSN);
}

// ---------------------------------------------------------------- out = vlad * rinv[nk] * finv[n]
__global__ void nv_scale_kernel(const float* __restrict__ vladb, const float* __restrict__ rinv,
                                const float* __restrict__ finv, float* __restrict__ out) {
  const size_t i = (size_t)blockIdx.x * 256 + threadIdx.x;
  const int nk = (int)(i >> 9);
  const int n  = (int)(i >> 15);
  out[i] = vladb[i] * rinv[nk] * finv[n];
}

extern "C" void kernel_launch(void* const* d_in, const int* in_sizes, int n_in,
                              void* d_out, int out_size, void* d_ws, size_t ws_size,
                              hipStream_t stream) {
  (void)in_sizes; (void)n_in; (void)out_size; (void)ws_size;
  const float* x      = (const float*)d_in[0];   // [32,512,32,32]
  const float* vocabs = (const float*)d_in[1];   // [64,512]
  float* out = (float*)d_out;                    // [32, 64*512]
  float* ws  = (float*)d_ws;

  // workspace layout (floats): ~12.2 MiB total
  float* bias    = ws;                           // 64
  float* invn    = bias + 64;                    // 32768
  float* assign  = invn + (NN * PP);             // 2,097,152
  float* vladb   = assign + (size_t)NN * KK * PP;// 1,048,576
  float* rinv    = vladb + (size_t)NN * KK * CC; // 2048
  float* partial = rinv + NN * KK;               // 2048
  float* finv    = partial + NN * KK;            // 32

  nv_bias_kernel          <<<KK, 256, 0, stream>>>(vocabs, bias);
  nv_invnorm_kernel       <<<(NN * PP) / 256, 256, 0, stream>>>(x, invn);
  nv_gemm1_softmax_kernel <<<NN * (PP / 32), 256, 0, stream>>>(x, vocabs, invn, bias, assign);
  nv_gemm2_vlad_kernel    <<<(NN * 4 * 32) / 8, 256, 0, stream>>>(x, assign, invn, vocabs, vladb);
  nv_rownorm_kernel       <<<NN * KK, 256, 0, stream>>>(vladb, rinv, partial);
  nv_finalnorm_kernel     <<<NN, 64, 0, stream>>>(partial, finv);
  nv_scale_kernel         <<<(NN * KK * CC) / 256, 256, 0, stream>>>(vladb, rinv, finv, out);
}